// HyperNet_3470333575391
// MI455X (gfx1250) — compile-verified
//
#include <hip/hip_runtime.h>

// ---------------------------------------------------------------------------
// HyperNet fused kernel for MI455X (gfx1250, wave32, WMMA f32 16x16x4).
//
// Never materializes w_all [8192 x 4096]:
//   Stage A:  S = X[8192x512] @ Wc[512x96]     (split-K: 2 waves per tile)
//   Stage B:  per-token epilogue in LDS -> P[16x68] = [outer(y4,h), y4]
//   Stage C:  Y = relu(P[8192x68] @ W2p[68x512]) (split-N: 16 tiles per wave)
//
// Weights are repacked by prep kernels into a K-PAIR-PACKED layout so each
// WMMA B fragment is a single 8-byte global_load_b64 (1:1 VMEM:WMMA issue in
// the hot loops).  Wc/W2p (<350KB) are L2-resident (192MB L2); HBM traffic is
// just X+Y = 34MB -> ~1.4us floor @ 23.3 TB/s.  ~1.38 GFLOP of fp32 WMMA.
// ---------------------------------------------------------------------------

typedef __attribute__((ext_vector_type(2))) float v2f;
typedef __attribute__((ext_vector_type(8))) float v8f;

#define EMBED 512
#define HID   16
#define NC    96      // stage-A combined N (16 h | 64 g | 4 c | 12 pad)
#define K2    68      // stage-C K (64 outer-product + 4 bias rows)
#define TOKENS 8192   // 8 * 1024
#define TILES_PER_BLK 2
#define WC_ELEMS  (EMBED * NC)   // 49152 floats  (24576 float2 pairs)
#define W2P_ELEMS (K2 * EMBED)   // 34816 floats  (17408 float2 pairs)

// ----------------------- prep kernels (pair-packed) ------------------------
// Layout: element (k, n) lives at pair p = k>>1, flat = (p*N + n)*2 + (k&1),
// so {row k, row k+1} for a given column n are one aligned float2.

__global__ void build_wc(const float* __restrict__ fc1_w,
                         const float* __restrict__ fc2_w,
                         const float* __restrict__ fc2_b,
                         float* __restrict__ Wcp) {
  int idx = blockIdx.x * blockDim.x + threadIdx.x;
  if (idx >= WC_ELEMS) return;
  int c = idx & 1, t = idx >> 1;
  int n = t % NC, p = t / NC;
  int e = p * 2 + c;                                // K index (embed dim)
  float v = 0.f;
  if (n < 16) {
    v = fc1_w[n * EMBED + e];                       // fc1_w[h][e]
  } else if (n < 80) {
    int j = (n - 16) >> 4, hh = (n - 16) & 15;
    v = fc2_w[(e * 4 + j) * HID + hh];              // W1 part of fc2_w
  } else if (n < 84) {
    v = fc2_b[e * 4 + (n - 80)];                    // W1 part of fc2_b
  }
  Wcp[idx] = v;
}

__global__ void build_w2p(const float* __restrict__ fc2_w,
                          const float* __restrict__ fc2_b,
                          float* __restrict__ W2pp) {
  int idx = blockIdx.x * blockDim.x + threadIdx.x;
  if (idx >= W2P_ELEMS) return;
  int c = idx & 1, t = idx >> 1;
  int e = t % EMBED, p = t / EMBED;
  int k = p * 2 + c;                                // K index (0..67)
  float v;
  if (k < 64) {
    int j = k >> 4, hh = k & 15;
    v = fc2_w[(2048 + j * EMBED + e) * HID + hh];   // W2 part of fc2_w
  } else {
    v = fc2_b[2048 + (k - 64) * EMBED + e];         // W2 part of fc2_b
  }
  W2pp[idx] = v;
}

// -------------------------------- main kernel ------------------------------
// Block = 128 threads = 4 waves = 2 token-tiles x 2 cooperating waves.
//   wave "half" 0/1: stage A contracts K half [0,256) / [256,512);
//                    stage C covers N tiles [0,16) / [16,32).

__global__ __launch_bounds__(128)
void hypernet_main(const float* __restrict__ X,
                   const float* __restrict__ fc1_b,
                   const float2* __restrict__ Wcp,
                   const float2* __restrict__ W2pp,
                   float* __restrict__ Y) {
  __shared__ __align__(16) float lds_S[TILES_PER_BLK][2][16][NC];  // 24 KB
  __shared__ __align__(16) float lds_P[TILES_PER_BLK][16][K2];     // 8.5 KB

  const int lane = threadIdx.x & 31;
  const int wv   = threadIdx.x >> 5;
  const int tile = wv >> 1;
  const int half = wv & 1;
  const int lc   = lane & 15;   // N column / M row index within 16-tile
  const int kd   = lane >> 4;   // lanes 16..31 hold K+2 (A/B) or M+8 (C/D)
  const int t0   = (blockIdx.x * TILES_PER_BLK + tile) * 16;

  // ---------------- Stage A: partial S = X[16 x 256] @ Wc[256 x 96] --------
  v8f acc0 = {0,0,0,0,0,0,0,0}, acc1 = {0,0,0,0,0,0,0,0},
      acc2 = {0,0,0,0,0,0,0,0}, acc3 = {0,0,0,0,0,0,0,0},
      acc4 = {0,0,0,0,0,0,0,0}, acc5 = {0,0,0,0,0,0,0,0};
  v8f* accs[6] = {&acc0, &acc1, &acc2, &acc3, &acc4, &acc5};

  const int k0 = half * (EMBED / 2);
  const float*  xrow  = X + (size_t)(t0 + lc) * EMBED + kd * 2;
  const float2* wcrow = Wcp + (size_t)kd * NC + lc;   // pair row for this lane
  for (int k = k0; k < k0 + EMBED / 2; k += 4) {
    // A fragment: A[M = lc, K = k + kd*2 + {0,1}]
    const float2 a2 = *(const float2*)(xrow + k);
    v2f A = {a2.x, a2.y};
    if (k + 64 < k0 + EMBED / 2) __builtin_prefetch(xrow + k + 64, 0, 1);
    const float2* wk = wcrow + (size_t)(k >> 1) * NC; // pair (k/2 + kd)
#pragma unroll
    for (int nt = 0; nt < 6; ++nt) {
      const float2 b2 = wk[nt * 16];                  // B[K=base..base+1, N]
      v2f B = {b2.x, b2.y};
      *accs[nt] = __builtin_amdgcn_wmma_f32_16x16x4_f32(
          false, A, false, B, (short)0, *accs[nt], false, false);
    }
  }
  // Spill partial S: D element (M = v + 8*kd, N = nt*16 + lc)
#pragma unroll
  for (int nt = 0; nt < 6; ++nt) {
#pragma unroll
    for (int v = 0; v < 8; ++v)
      lds_S[tile][half][v + kd * 8][nt * 16 + lc] = (*accs[nt])[v];
  }
  __syncthreads();

  // ---------------- Stage B: per-token tiny epilogue -> P[16x68] -----------
  if (half == 0 && lane < 16) {
    const int m = lane;
    float h[16];
#pragma unroll
    for (int i = 0; i < 16; ++i)
      h[i] = fmaxf(lds_S[tile][0][m][i] + lds_S[tile][1][m][i] + fc1_b[i], 0.f);
#pragma unroll
    for (int j = 0; j < 4; ++j) {
      float s = lds_S[tile][0][m][80 + j] + lds_S[tile][1][m][80 + j];
#pragma unroll
      for (int i = 0; i < 16; ++i)
        s += (lds_S[tile][0][m][16 + j * 16 + i] +
              lds_S[tile][1][m][16 + j * 16 + i]) * h[i];
      float y4 = fmaxf(s, 0.f);
#pragma unroll
      for (int i = 0; i < 16; ++i)
        lds_P[tile][m][j * 16 + i] = y4 * h[i];      // outer(y4, h)
      lds_P[tile][m][64 + j] = y4;                   // bias rows
    }
  }
  __syncthreads();

  // ---------------- Stage C: Y = relu(P[16x68] @ W2p[68x512]) --------------
  // Preload all 17 A fragments from LDS once; reuse across 16 N-tiles.
  v2f Af[17];
#pragma unroll
  for (int ks = 0; ks < 17; ++ks) {
    const float2 a2 = *(const float2*)&lds_P[tile][lc][ks * 4 + kd * 2];
    Af[ks] = (v2f){a2.x, a2.y};
  }
  const float2* w2row = W2pp + (size_t)kd * EMBED + lc;  // pair row, this lane
  float* yrow = Y + (size_t)t0 * EMBED;
  for (int nt = half * 16; nt < half * 16 + 16; ++nt) {
    v8f c = {0,0,0,0,0,0,0,0};
#pragma unroll
    for (int ks = 0; ks < 17; ++ks) {
      // pair index = ks*2 + kd -> single b64 per fragment
      const float2 b2 = w2row[(size_t)(ks * 2) * EMBED + nt * 16];
      v2f B = {b2.x, b2.y};
      c = __builtin_amdgcn_wmma_f32_16x16x4_f32(
          false, Af[ks], false, B, (short)0, c, false, false);
    }
#pragma unroll
    for (int v = 0; v < 8; ++v)
      yrow[(size_t)(v + kd * 8) * EMBED + nt * 16 + lc] = fmaxf(c[v], 0.f);
  }
}

// ------------------------------- launcher ----------------------------------

extern "C" void kernel_launch(void* const* d_in, const int* in_sizes, int n_in,
                              void* d_out, int out_size, void* d_ws, size_t ws_size,
                              hipStream_t stream) {
  const float* x     = (const float*)d_in[0];
  const float* fc1_w = (const float*)d_in[1];
  const float* fc1_b = (const float*)d_in[2];
  const float* fc2_w = (const float*)d_in[3];
  const float* fc2_b = (const float*)d_in[4];
  float* Wcp  = (float*)d_ws;
  float* W2pp = Wcp + WC_ELEMS;

  build_wc <<<(WC_ELEMS  + 255) / 256, 256, 0, stream>>>(fc1_w, fc2_w, fc2_b, Wcp);
  build_w2p<<<(W2P_ELEMS + 255) / 256, 256, 0, stream>>>(fc2_w, fc2_b, W2pp);

  hypernet_main<<<TOKENS / (16 * TILES_PER_BLK), 128, 0, stream>>>(
      x, fc1_b, (const float2*)Wcp, (const float2*)W2pp, (float*)d_out);
}